// Encoder_49984829391224
// MI455X (gfx1250) — compile-verified
//
#include <hip/hip_runtime.h>
#include <cstdint>

#define WPB 8           // waves per block in the aggregation kernel
#define DIM 128
#define PIPE 4          // async rows in flight per wave (depth-4 LDS pipeline)

// ---------------------------------------------------------------------------
// CDNA5 async global->LDS copy: each lane copies 16B; a full wave32 moves one
// 512B feature row (128 f32) in a single instruction, tracked on ASYNCcnt.
// ---------------------------------------------------------------------------
__device__ __forceinline__ void async_copy16(const float* g, unsigned lds_off) {
  asm volatile("global_load_async_to_lds_b128 %0, %1, off"
               :: "v"(lds_off), "v"((unsigned long long)(uintptr_t)g)
               : "memory");
}

// Reissue into a buffer we just ds_load'ed from: drain DScnt first so the
// in-flight async write can never land before our LDS read executed.
__device__ __forceinline__ void issue_row_safe(const float* g, unsigned lds_off) {
  asm volatile("s_wait_dscnt 0x0\n\t"
               "global_load_async_to_lds_b128 %0, %1, off"
               :: "v"(lds_off), "v"((unsigned long long)(uintptr_t)g)
               : "memory");
}

__device__ __forceinline__ void wait_async_le3() {
  asm volatile("s_wait_asynccnt 3" ::: "memory");
}
__device__ __forceinline__ void wait_async_0() {
  asm volatile("s_wait_asynccnt 0" ::: "memory");
}

// ---------------------------------------------------------------------------
// K0: init deg (self loop => 1.0), zero counters
// ---------------------------------------------------------------------------
__global__ void k_init(float* __restrict__ deg, int* __restrict__ cnt,
                       int* __restrict__ fill, int n) {
  int i = blockIdx.x * blockDim.x + threadIdx.x;
  if (i < n) { deg[i] = 1.0f; cnt[i] = 0; fill[i] = 0; }
}

// ---------------------------------------------------------------------------
// K1: degree by src (float adds of 1.0 are exact & order-independent) +
//     dst histogram for the CSR build
// ---------------------------------------------------------------------------
__global__ void k_count(const int* __restrict__ edges, float* __restrict__ deg,
                        int* __restrict__ cnt, int e) {
  int i = blockIdx.x * blockDim.x + threadIdx.x;
  if (i < e) {
    int s = edges[2 * i];
    int d = edges[2 * i + 1];
    atomicAdd(&deg[s], 1.0f);
    atomicAdd(&cnt[d], 1);
  }
}

// ---------------------------------------------------------------------------
// K2a: per-1024-tile reduction of cnt -> bsum[block]; rsqrt of deg fused in
// (deg is final after K1; deg >= 1 so no zero guard needed)
// ---------------------------------------------------------------------------
__global__ void k_scan_partial(const int* __restrict__ cnt, int* __restrict__ bsum,
                               float* __restrict__ deg, int n) {
  __shared__ int sdata[1024];
  const int t = threadIdx.x;
  const int i = blockIdx.x * 1024 + t;
  if (i < n) deg[i] = rsqrtf(deg[i]);
  sdata[t] = (i < n) ? cnt[i] : 0;
  __syncthreads();
  for (int off = 512; off > 0; off >>= 1) {
    if (t < off) sdata[t] += sdata[t + off];
    __syncthreads();
  }
  if (t == 0) bsum[blockIdx.x] = sdata[0];
}

// ---------------------------------------------------------------------------
// K2b: exclusive scan of block sums in place (single block; nb <= 1024)
// ---------------------------------------------------------------------------
__global__ void k_scan_bsum(int* __restrict__ bsum, int nb) {
  __shared__ int sdata[1024];
  const int t = threadIdx.x;
  const int v = (t < nb) ? bsum[t] : 0;
  sdata[t] = v;
  __syncthreads();
  for (int off = 1; off < 1024; off <<= 1) {
    int add = (t >= off) ? sdata[t - off] : 0;
    __syncthreads();
    sdata[t] += add;
    __syncthreads();
  }
  if (t < nb) bsum[t] = sdata[t] - v;   // exclusive
}

// ---------------------------------------------------------------------------
// K2c: per-tile exclusive scan + block offset -> rowptr
// ---------------------------------------------------------------------------
__global__ void k_scan_final(const int* __restrict__ cnt, const int* __restrict__ bsum,
                             int* __restrict__ rowptr, int n) {
  __shared__ int sdata[1024];
  const int t = threadIdx.x;
  const int i = blockIdx.x * 1024 + t;
  const int v = (i < n) ? cnt[i] : 0;
  sdata[t] = v;
  __syncthreads();
  for (int off = 1; off < 1024; off <<= 1) {
    int add = (t >= off) ? sdata[t - off] : 0;
    __syncthreads();
    sdata[t] += add;
    __syncthreads();
  }
  if (i < n) rowptr[i] = bsum[blockIdx.x] + sdata[t] - v;
}

// ---------------------------------------------------------------------------
// K3: bucket src ids by dst (CSR fill) — 1.6M int atomics instead of 436M
// f32 atomics in feature space.
// ---------------------------------------------------------------------------
__global__ void k_fill(const int* __restrict__ edges, const int* __restrict__ rowptr,
                       int* __restrict__ fill, int* __restrict__ col, int e) {
  int i = blockIdx.x * blockDim.x + threadIdx.x;
  if (i < e) {
    int s = edges[2 * i];
    int d = edges[2 * i + 1];
    int slot = atomicAdd(&fill[d], 1);
    col[rowptr[d] + slot] = s;
  }
}

// ---------------------------------------------------------------------------
// K4: pull-mode GCN layer. One wave32 per destination node; lane owns 4
// consecutive features (float4). Neighbor rows stream through a per-wave
// 4-deep LDS ring via async global->LDS b128 copies (3 rows in flight,
// s_wait_asynccnt 3 steady state). The dependent scalar chain
// col[k+1] -> dinv[col[k+1]] is software-pipelined one iteration ahead.
// Self-loop, bias, ReLU fused. No atomics; one store per output row.
// ---------------------------------------------------------------------------
template <bool RELU>
__global__ void __launch_bounds__(WPB * 32)
k_aggregate(const float* __restrict__ x, const float* __restrict__ w,
            const float* __restrict__ b, const float* __restrict__ dinv,
            const int* __restrict__ rowptr, const int* __restrict__ cnt,
            const int* __restrict__ col, float* __restrict__ out, int n) {
  __shared__ __align__(16) float rowbuf[WPB][PIPE][DIM];
  const int lane = threadIdx.x & 31;
  const int wv   = threadIdx.x >> 5;
  const int node = blockIdx.x * WPB + wv;
  if (node >= n) return;                 // uniform per wave

  const int lo4 = lane * 4;
  const float di = dinv[node];
  const float4 wvec = *(const float4*)(w + lo4);
  const float4 xv   = *(const float4*)(x + (size_t)node * DIM + lo4);

  // self-loop: norm = dinv[i]^2
  const float sl = di * di;
  float ax = sl * xv.x * wvec.x;
  float ay = sl * xv.y * wvec.y;
  float az = sl * xv.z * wvec.z;
  float aw = sl * xv.w * wvec.w;

  const int beg = rowptr[node];
  const int m   = cnt[node];

  // contiguous per-wave ring: buffer j is at base + j*512 bytes
  const unsigned b0 = (unsigned)(uintptr_t)&rowbuf[wv][0][lo4];

  // prologue: fill the ring (col entries share one cacheline -> cheap)
  #pragma unroll
  for (int j = 0; j < PIPE; ++j) {
    if (j < m) {
      int sj = col[beg + j];
      async_copy16(x + (size_t)sj * DIM + lo4, b0 + (unsigned)(j << 9));
    }
  }

  float dv_cur = (m > 0) ? dinv[col[beg]] : 0.0f;
  for (int k = 0; k < m; ++k) {
    // pipeline the col -> dinv chain one iteration ahead
    float dv_nxt = (k + 1 < m) ? dinv[col[beg + k + 1]] : 0.0f;

    const bool refill = (k + PIPE < m);
    if (refill) wait_async_le3();        // row k landed, 3 rows in flight
    else        wait_async_0();          // tail: drain (no-op after first)

    const float4 v = *(const float4*)(&rowbuf[wv][k & (PIPE - 1)][lo4]);
    if (refill) {
      int s2 = col[beg + k + PIPE];
      // s_wait_dscnt 0 inside: our ds_load above must finish before the
      // async write can reuse this ring slot
      issue_row_safe(x + (size_t)s2 * DIM + lo4,
                     b0 + (unsigned)((k & (PIPE - 1)) << 9));
    }

    const float ns = dv_cur * di;
    ax += ns * v.x * wvec.x;
    ay += ns * v.y * wvec.y;
    az += ns * v.z * wvec.z;
    aw += ns * v.w * wvec.w;
    dv_cur = dv_nxt;
  }

  const float4 bv = *(const float4*)(b + lo4);
  ax += bv.x; ay += bv.y; az += bv.z; aw += bv.w;
  if (RELU) {
    ax = fmaxf(ax, 0.0f); ay = fmaxf(ay, 0.0f);
    az = fmaxf(az, 0.0f); aw = fmaxf(aw, 0.0f);
  }
  float4 o; o.x = ax; o.y = ay; o.z = az; o.w = aw;
  *(float4*)(out + (size_t)node * DIM + lo4) = o;
}

// ---------------------------------------------------------------------------
// Launcher
// inputs: edges(int32 E*2), x(f32 N*128), w1(128), b1(128), w2(128), b2(128)
// ---------------------------------------------------------------------------
extern "C" void kernel_launch(void* const* d_in, const int* in_sizes, int n_in,
                              void* d_out, int out_size, void* d_ws, size_t ws_size,
                              hipStream_t stream) {
  const int*   edges = (const int*)d_in[0];
  const float* x     = (const float*)d_in[1];
  const float* w1    = (const float*)d_in[2];
  const float* b1    = (const float*)d_in[3];
  const float* w2    = (const float*)d_in[4];
  const float* b2    = (const float*)d_in[5];
  float* out = (float*)d_out;

  const int E  = in_sizes[0] / 2;
  const int N  = in_sizes[1] / DIM;
  const int NB = (N + 1023) / 1024;       // scan tiles (98 for N=100k)

  // workspace carve-out (256B aligned): ~59 MB total
  uintptr_t p = (uintptr_t)d_ws;
  auto take = [&](size_t bytes) -> void* {
    void* r = (void*)p;
    p = (p + bytes + 255) & ~(uintptr_t)255;
    return r;
  };
  float* deg    = (float*)take((size_t)N * 4);        // becomes dinv in place
  int*   cnt    = (int*)  take((size_t)N * 4);
  int*   rowptr = (int*)  take((size_t)N * 4);
  int*   fill   = (int*)  take((size_t)N * 4);
  int*   bsum   = (int*)  take((size_t)NB * 4);
  int*   col    = (int*)  take((size_t)E * 4);
  float* h1     = (float*)take((size_t)N * DIM * 4);
  (void)ws_size; (void)n_in; (void)out_size;

  const int tb = 256;
  k_init        <<<(N + tb - 1) / tb, tb, 0, stream>>>(deg, cnt, fill, N);
  k_count       <<<(E + tb - 1) / tb, tb, 0, stream>>>(edges, deg, cnt, E);
  k_scan_partial<<<NB, 1024, 0, stream>>>(cnt, bsum, deg, N);   // + rsqrt fused
  k_scan_bsum   <<<1, 1024, 0, stream>>>(bsum, NB);
  k_scan_final  <<<NB, 1024, 0, stream>>>(cnt, bsum, rowptr, N);
  k_fill        <<<(E + tb - 1) / tb, tb, 0, stream>>>(edges, rowptr, fill, col, E);

  const int ablocks = (N + WPB - 1) / WPB;
  k_aggregate<true ><<<ablocks, WPB * 32, 0, stream>>>(x,  w1, b1, deg, rowptr, cnt, col, h1,  N);
  k_aggregate<false><<<ablocks, WPB * 32, 0, stream>>>(h1, w2, b2, deg, rowptr, cnt, col, out, N);
}